// Head_10144712753551
// MI455X (gfx1250) — compile-verified
//
#include <hip/hip_runtime.h>

// ---------------------------------------------------------------------------
// RefinedBSA head for MI455X (gfx1250, wave32).
// One workgroup (256 threads = 8 waves, 2 per SIMD32) per batch element.
// LDS (~197 KB): score plane overlaps the dead x / W^T staging region.
// fp32 WMMA (v_wmma_f32_16x16x4_f32) for projections and q.k^T; W staged
// pair-interleaved so every WMMA fragment is a single ds_load_b64.
// Sparse top-8 gather replaces the dense attn@v GEMM (16x fewer FLOPs).
// ---------------------------------------------------------------------------

typedef __attribute__((ext_vector_type(2))) float v2f;
typedef __attribute__((ext_vector_type(4))) float v4f;
typedef __attribute__((ext_vector_type(8))) float v8f;

namespace {
constexpr int   T      = 128;       // sequence length
constexpr int   H      = 64;        // head size == n_embd
constexpr int   TOPK   = 8;
constexpr float ALPHA  = 0.1f;
constexpr float SCALE  = 0.125f;    // 1/sqrt(64)

constexpr int SX = 72;              // padded row stride for X/Q/K/V (floats)
constexpr int SF = 136;             // padded row stride for score plane (floats)
constexpr int SW = 130;             // pair-interleaved W stride: 2*64 + 2 pad

// Region A (reused): [ X (128x72) | Wp (3 x 32x130) ]  then  [ F (128x136) ]
constexpr int X_OFF  = 0;                    // live: phase A..B
constexpr int WT_OFF = X_OFF + T * SX;       // 9216, live: phase A..B
constexpr int WSZ    = 32 * SW;              // 4160 floats per matrix
constexpr int F_OFF  = 0;                    // live: phase C..D (overlaps X/Wp)
constexpr int REGA   = WT_OFF + 3 * WSZ;     // 21696 >= 128*136 = 17408  (F fits)
// Region B: q,k,v contiguous
constexpr int Q_OFF  = REGA;                 // 21696
constexpr int SMEM_FLOATS = Q_OFF + 3 * T * SX;   // 49344 floats = 197376 bytes
}

__global__ __launch_bounds__(256)
void bsa_head_kernel(const float* __restrict__ x,
                     const float* __restrict__ Wk,
                     const float* __restrict__ Wq,
                     const float* __restrict__ Wv,
                     const float* __restrict__ gamma,
                     float* __restrict__ out)
{
    extern __shared__ float sm[];

    const int b     = blockIdx.x;
    const int tid   = threadIdx.x;
    const int lane  = tid & 31;
    const int lhalf = lane & 15;    // N / M index within half-wave
    const int hsel  = lane >> 4;    // half-wave select (K pair / M+8)
    const int wave  = tid >> 5;     // 0..7

    // ---------------- Phase A: stage x[b] and pair-interleaved W ----------------
    const float* xb = x + (size_t)b * (T * H);
    for (int i = tid; i < T * H / 4; i += 256) {
        const int row = i >> 4;     // 16 float4 per row
        const int c4  = i & 15;
        v4f val = *(const v4f*)(xb + row * H + c4 * 4);
        *(v4f*)(sm + X_OFF + row * SX + c4 * 4) = val;
    }
    // Wp slots: 0 -> Wq (q), 1 -> Wk (k), 2 -> Wv (v).
    // B = W^T:  B[d][h] = W[h][d];  Wp[d>>1][h*2 + (d&1)] = W[h][d]
    for (int p = 0; p < 3; ++p) {
        const float* W = (p == 0) ? Wq : (p == 1) ? Wk : Wv;
        float* wp = sm + WT_OFF + p * WSZ;
        for (int i = tid; i < H * H; i += 256) {
            const int h = i >> 6;
            const int d = i & 63;
            wp[(d >> 1) * SW + h * 2 + (d & 1)] = W[h * H + d];
        }
    }
    __syncthreads();

    // ---------------- Phase B: q/k/v = x @ W^T via fp32 WMMA ----------------
    // 96 tiles: proj(3) x mtile(8) x ntile(4); round-robin over 8 waves.
    for (int tile = wave; tile < 96; tile += 8) {
        const int p   = tile >> 5;
        const int rem = tile & 31;
        const int m0  = (rem >> 2) * 16;
        const int n0  = (rem & 3) * 16;
        const float* A  = sm + X_OFF;
        const float* Wp = sm + WT_OFF + p * WSZ;
        float*       D  = sm + Q_OFF + p * (T * SX);

        v8f acc = {};
#pragma unroll
        for (int kk = 0; kk < H; kk += 4) {
            const int kIdx = kk + 2 * hsel;   // even
            // A 16x4 frag: lane's two consecutive K values of its row
            v2f a  = *(const v2f*)(A + (m0 + lhalf) * SX + kIdx);
            // B 4x16 frag: {B[kIdx][n], B[kIdx+1][n]} contiguous in Wp
            v2f bb = *(const v2f*)(Wp + (kIdx >> 1) * SW + (n0 + lhalf) * 2);
            acc = __builtin_amdgcn_wmma_f32_16x16x4_f32(
                false, a, false, bb, (short)0, acc, false, false);
        }
#pragma unroll
        for (int r = 0; r < 8; ++r)
            D[(m0 + r + 8 * hsel) * SX + n0 + lhalf] = acc[r];
    }
    __syncthreads();

    // ------- Phase C: f = relu(scale * q k^T, causal) * decay via WMMA -------
    // (F overlaps the now-dead X/Wp region.)
    const float* Qs = sm + Q_OFF;
    const float* Ks = sm + Q_OFF + T * SX;
    for (int tile = wave; tile < 64; tile += 8) {
        const int m0 = (tile >> 3) * 16;
        const int n0 = (tile & 7) * 16;

        v8f acc = {};
#pragma unroll
        for (int kk = 0; kk < H; kk += 4) {
            const int kIdx = kk + 2 * hsel;
            v2f a  = *(const v2f*)(Qs + (m0 + lhalf) * SX + kIdx);
            // B = k^T: B[kIdx][j] = k[j][kIdx]; consecutive columns of row j
            v2f bb = *(const v2f*)(Ks + (n0 + lhalf) * SX + kIdx);
            acc = __builtin_amdgcn_wmma_f32_16x16x4_f32(
                false, a, false, bb, (short)0, acc, false, false);
        }
#pragma unroll
        for (int r = 0; r < 8; ++r) {
            const int i = m0 + r + 8 * hsel;
            const int j = n0 + lhalf;
            float fv = 0.0f;
            if (j <= i) {
                const float decay = 1.0f - (ALPHA / 128.0f) * (float)(i - j);
                fv = fmaxf(acc[r] * SCALE, 0.0f) * decay;
            }
            sm[F_OFF + i * SF + j] = fv;   // masked entries: relu(-inf)=0
        }
    }
    __syncthreads();

    // ------ Phase D: per-row stats, top-8, int8 quant, sparse attn @ v ------
    // Two threads per row (lanes differing in bit 0); combine via shfl_xor(1).
    {
        const int row  = tid >> 1;          // 0..127
        const int half = tid & 1;
        const int j0   = half * 64;
        float* rf = sm + F_OFF + row * SF;

        float m = 0.0f, s1 = 0.0f, s2 = 0.0f;   // f >= 0 so max init 0 is exact
        for (int j = j0; j < j0 + 64; ++j) {
            const float fv = rf[j];
            m  = fmaxf(m, fv);
            s1 += fv;
            s2 += fv * fv;
        }
        m   = fmaxf(m, __shfl_xor(m, 1));
        s1 += __shfl_xor(s1, 1);
        s2 += __shfl_xor(s2, 1);

        const float var   = (s2 - s1 * s1 * (1.0f / T)) * (1.0f / (T - 1)); // ddof=1
        const float sigma = sqrtf(fmaxf(var, 0.0f));
        const float denom = fmaxf(m, sigma) + 1e-6f;
        const float g     = gamma[0];

        float w[TOPK];
        int   idx[TOPK];
#pragma unroll
        for (int sel = 0; sel < TOPK; ++sel) {
            float best = -1.0f;
            int   bj   = 0;
            for (int j = j0; j < j0 + 64; ++j) {
                const float fv = rf[j];
                if (fv > best) { best = fv; bj = j; }
            }
            // combine the two halves; lower index wins ties (first occurrence)
            const float pb = __shfl_xor(best, 1);
            const int   pj = __shfl_xor(bj, 1);
            if (pb > best || (pb == best && pj < bj)) { best = pb; bj = pj; }
            rf[bj] = -2.0f;                      // both lanes store same value

            float qv = floorf(255.0f * best / denom);
            qv = fminf(fmaxf(qv, 0.0f), 255.0f);
            if (qv > 127.0f) qv -= 256.0f;       // emulate int8 wraparound
            w[sel]   = qv / g;
            idx[sel] = bj;
        }

        // out[row] = sum_sel w[sel] * v[idx[sel]]; each thread owns 32 channels
        v4f acc[8];
#pragma unroll
        for (int c = 0; c < 8; ++c) acc[c] = (v4f){0.0f, 0.0f, 0.0f, 0.0f};
        const float* Vs = sm + Q_OFF + 2 * T * SX;
#pragma unroll
        for (int sel = 0; sel < TOPK; ++sel) {
            const float ww = w[sel];
            const v4f* vr = (const v4f*)(Vs + idx[sel] * SX) + half * 8;
#pragma unroll
            for (int c = 0; c < 8; ++c) acc[c] += ww * vr[c];
        }
        float* op = out + ((size_t)b * T + row) * H + half * 32;
#pragma unroll
        for (int c = 0; c < 8; ++c) *(v4f*)(op + c * 4) = acc[c];
    }
}

extern "C" void kernel_launch(void* const* d_in, const int* in_sizes, int n_in,
                              void* d_out, int out_size, void* d_ws, size_t ws_size,
                              hipStream_t stream) {
    const float* x     = (const float*)d_in[0];
    const float* Wk    = (const float*)d_in[1];
    const float* Wq    = (const float*)d_in[2];
    const float* Wv    = (const float*)d_in[3];
    const float* gamma = (const float*)d_in[4];
    float* out = (float*)d_out;

    const int B = in_sizes[0] / (T * H);   // 4096
    const size_t shmem = (size_t)SMEM_FLOATS * sizeof(float);  // 197376 B

    bsa_head_kernel<<<B, 256, shmem, stream>>>(x, Wk, Wq, Wv, gamma, out);
}